// ResGATEAUBlock_17506286698854
// MI455X (gfx1250) — compile-verified
//
#include <hip/hip_runtime.h>
#include <hip/hip_bf16.h>
#include <math.h>

// ---------------------------------------------------------------------------
// ResGATEAU block (2x GAT layer + LN + BN + residual) for MI455X (gfx1250).
// fp32 throughout (memory-bound workload; no precision downgrade needed).
// Dense 256x256 GEMMs use V_WMMA_F32_16X16X4_F32; A-tile staging uses the
// gfx1250 async global->LDS path (GLOBAL_LOAD_ASYNC_TO_LDS_B128 + asynccnt).
// ---------------------------------------------------------------------------

#define NN 20000
#define EE 320000
#define ETOT (EE + NN)
#define DD 256
#define HH 8
#define CC 32
#define EDIM 16
#define SELF_LOOP_TYPE 6
#define NEG_SLOPE 0.2f
#define EPS_LN 1e-5f
#define EPS_BN 1e-5f

typedef __attribute__((ext_vector_type(2))) float v2f;
typedef __attribute__((ext_vector_type(8))) float v8f;
typedef __attribute__((ext_vector_type(4))) int v4i;
typedef __attribute__((address_space(1))) v4i* gv4i_p;   // global 16B vector ptr
typedef __attribute__((address_space(3))) v4i* lv4i_p;   // LDS    16B vector ptr

#if __has_builtin(__builtin_amdgcn_global_load_async_to_lds_b128)
#define HAVE_ASYNC_LDS 1
#endif

// ------------------------- helpers -----------------------------------------
__device__ __forceinline__ unsigned f2ord(float f) {
    unsigned u = __float_as_uint(f);
    return (u & 0x80000000u) ? ~u : (u | 0x80000000u);
}
__device__ __forceinline__ float ord2f(unsigned u) {
    return (u & 0x80000000u) ? __uint_as_float(u & 0x7fffffffu)
                             : __uint_as_float(~u);
}

// ------------------------- zero fill ----------------------------------------
__global__ void k_zero(float* __restrict__ p, long n) {
    long i = (long)blockIdx.x * blockDim.x + threadIdx.x;
    if (i < n) p[i] = 0.0f;
}

// ------------- fold attention vectors into weight matrices ------------------
// vWd[d,h] = sum_c Wd[d, h*32+c] * a_s[h,c]     (for a_i, paired with dst)
// vWs[d,h] = sum_c Ws[d, h*32+c] * a_d[h,c]     (for a_j, paired with src)
// Me [k,h] = sum_c We[k, h*32+c] * ae[h,c]
// tt [t,h] = sum_c te[t, h*32+c] * ae[h,c]
__global__ void k_prep(const float* __restrict__ Wd, const float* __restrict__ as_,
                       const float* __restrict__ Ws, const float* __restrict__ ad_,
                       const float* __restrict__ We, const float* __restrict__ ae_,
                       const float* __restrict__ te,
                       float* __restrict__ vWd, float* __restrict__ vWs,
                       float* __restrict__ Me, float* __restrict__ tt) {
    int t = blockIdx.x * blockDim.x + threadIdx.x;
    if (t < 2048) {
        int d = t >> 3, h = t & 7;
        float s = 0.f;
        for (int c = 0; c < CC; ++c) s += Wd[d * DD + h * CC + c] * as_[h * CC + c];
        vWd[t] = s;
    } else if (t < 4096) {
        int t2 = t - 2048, d = t2 >> 3, h = t2 & 7;
        float s = 0.f;
        for (int c = 0; c < CC; ++c) s += Ws[d * DD + h * CC + c] * ad_[h * CC + c];
        vWs[t2] = s;
    } else if (t < 4096 + EDIM * HH) {
        int t2 = t - 4096, k = t2 >> 3, h = t2 & 7;
        float s = 0.f;
        for (int c = 0; c < CC; ++c) s += We[k * DD + h * CC + c] * ae_[h * CC + c];
        Me[t2] = s;
    } else if (t < 4096 + EDIM * HH + 7 * HH) {
        int t2 = t - 4096 - EDIM * HH, ty = t2 >> 3, h = t2 & 7;
        float s = 0.f;
        for (int c = 0; c < CC; ++c) s += te[ty * DD + h * CC + c] * ae_[h * CC + c];
        tt[t2] = s;
    }
}

// ------------------------- WMMA f32 GEMM ------------------------------------
// C[Nrows,256] = A[Nrows,256] @ B[256,256], fp32, V_WMMA_F32_16X16X4_F32.
// One block = 256 threads = 8 waves; block owns a 16-row strip; each wave
// owns two 16-col tiles. A tile staged in LDS via async global->LDS copy;
// B streamed from global (L2-resident 256 KB weight).
__global__ __launch_bounds__(256) void k_gemm256(const float* __restrict__ A,
                                                 const float* __restrict__ B,
                                                 float* __restrict__ C) {
    __shared__ float As[16][DD];  // 16 KB
    const int m0 = blockIdx.x * 16;
    const int tid = threadIdx.x;

    // stage A tile: 16x256 floats = 1024 x 16B transfers, 4 per thread
#ifdef HAVE_ASYNC_LDS
    for (int i = tid; i < 16 * 64; i += 256) {
        int r = i >> 6;
        int c4 = i & 63;
        const float* g = A + (size_t)(m0 + r) * DD + c4 * 4;
        __builtin_amdgcn_global_load_async_to_lds_b128(
            (gv4i_p)(const_cast<float*>(g)),
            (lv4i_p)&As[r][c4 * 4],
            0, 0);
    }
#if __has_builtin(__builtin_amdgcn_s_wait_asynccnt)
    __builtin_amdgcn_s_wait_asynccnt(0);
#else
    asm volatile("s_wait_asynccnt 0x0" ::: "memory");
#endif
    __syncthreads();
#else
    for (int i = tid; i < 16 * 64; i += 256) {
        int r = i >> 6;
        int c4 = i & 63;
        float4 v = ((const float4*)(A + (size_t)(m0 + r) * DD))[c4];
        *(float4*)&As[r][c4 * 4] = v;
    }
    __syncthreads();
#endif

    const int wave = tid >> 5;
    const int lane = tid & 31;
    const int half = lane >> 4;   // K-half selector for A/B fragments
    const int l16  = lane & 15;
    const int n0   = wave * 32;   // 2 x 16 output columns per wave

    v8f acc0 = {};
    v8f acc1 = {};
    for (int k = 0; k < DD; k += 4) {
        // A frag: lanes 0-15 -> K = k+0,k+1 ; lanes 16-31 -> K = k+2,k+3
        v2f a;
        a.x = As[l16][k + 2 * half + 0];
        a.y = As[l16][k + 2 * half + 1];
        const float* Bk0 = B + (size_t)(k + 2 * half) * DD;
        v2f b0, b1;
        b0.x = Bk0[n0 + l16];
        b0.y = Bk0[DD + n0 + l16];
        b1.x = Bk0[n0 + 16 + l16];
        b1.y = Bk0[DD + n0 + 16 + l16];
        acc0 = __builtin_amdgcn_wmma_f32_16x16x4_f32(false, a, false, b0,
                                                     (short)0, acc0, false, false);
        acc1 = __builtin_amdgcn_wmma_f32_16x16x4_f32(false, a, false, b1,
                                                     (short)0, acc1, false, false);
    }
    // D layout: VGPR r -> lanes 0-15: M=r, lanes 16-31: M=r+8 ; N=lane&15
    for (int r = 0; r < 8; ++r) {
        int m = m0 + r + 8 * half;
        int n = n0 + l16;
        C[(size_t)m * DD + n]      = acc0[r];
        C[(size_t)m * DD + n + 16] = acc1[r];
    }
}

// ---------------- per-node attention logits (warp per node) -----------------
__global__ __launch_bounds__(256) void k_node_logits(const float* __restrict__ X,
                                                     const float* __restrict__ vWd,
                                                     const float* __restrict__ vWs,
                                                     float* __restrict__ a_i,
                                                     float* __restrict__ a_j) {
    __shared__ float sWd[DD * HH];
    __shared__ float sWs[DD * HH];
    for (int i = threadIdx.x; i < DD * HH; i += blockDim.x) {
        sWd[i] = vWd[i];
        sWs[i] = vWs[i];
    }
    __syncthreads();
    int warp = (int)((blockIdx.x * blockDim.x + threadIdx.x) >> 5);
    int lane = threadIdx.x & 31;
    if (warp >= NN) return;
    float pi[HH] = {0}, pj[HH] = {0};
    for (int j = 0; j < 8; ++j) {
        int d = lane + 32 * j;
        float xv = X[(size_t)warp * DD + d];
        for (int h = 0; h < HH; ++h) {
            pi[h] += xv * sWd[d * HH + h];
            pj[h] += xv * sWs[d * HH + h];
        }
    }
    for (int h = 0; h < HH; ++h) {
        float si = pi[h], sj = pj[h];
        for (int o = 16; o > 0; o >>= 1) {
            si += __shfl_xor(si, o, 32);
            sj += __shfl_xor(sj, o, 32);
        }
        if (lane == 0) {
            a_i[(size_t)warp * HH + h] = si;
            a_j[(size_t)warp * HH + h] = sj;
        }
    }
}

// ---------------- edge logits + leaky relu + segment max --------------------
__global__ void k_edge_alpha(const int* __restrict__ src, const int* __restrict__ dst,
                             const float* __restrict__ ea, const int* __restrict__ et,
                             const float* __restrict__ a_i, const float* __restrict__ a_j,
                             const float* __restrict__ Me, const float* __restrict__ tt,
                             float* __restrict__ alpha, unsigned* __restrict__ mbuf) {
    int e = blockIdx.x * blockDim.x + threadIdx.x;
    if (e >= ETOT) return;
    int s_, d_, t_;
    float eav[EDIM];
    if (e < EE) {
        s_ = src[e];
        d_ = dst[e];
        t_ = et[e];
        for (int k = 0; k < EDIM; ++k) eav[k] = ea[(size_t)e * EDIM + k];
    } else {
        int n = e - EE;
        s_ = n; d_ = n; t_ = SELF_LOOP_TYPE;
        for (int k = 0; k < EDIM; ++k) eav[k] = 1.0f;
    }
    for (int h = 0; h < HH; ++h) {
        float al = a_i[(size_t)d_ * HH + h] + a_j[(size_t)s_ * HH + h] + tt[t_ * HH + h];
        for (int k = 0; k < EDIM; ++k) al += eav[k] * Me[k * HH + h];
        al = (al >= 0.f) ? al : NEG_SLOPE * al;
        alpha[(size_t)e * HH + h] = al;
        atomicMax(&mbuf[(size_t)d_ * HH + h], f2ord(al));
    }
}

// ---------------- exp(alpha - max) + segment sum ----------------------------
__global__ void k_edge_exp(const int* __restrict__ dst, float* __restrict__ alpha,
                           const unsigned* __restrict__ mbuf, float* __restrict__ sbuf) {
    long idx = (long)blockIdx.x * blockDim.x + threadIdx.x;
    if (idx >= (long)ETOT * HH) return;
    int e = (int)(idx >> 3);
    int h = (int)(idx & 7);
    int d_ = (e < EE) ? dst[e] : (e - EE);
    float w = alpha[idx];
    float mm = ord2f(mbuf[(size_t)d_ * HH + h]);
    float ex = expf(w - mm);
    alpha[idx] = ex;
    atomicAdd(&sbuf[(size_t)d_ * HH + h], ex);
}

// ---------------- message gather + scatter-add (warp per edge) --------------
__global__ __launch_bounds__(256) void k_scatter(const int* __restrict__ src,
                                                 const int* __restrict__ dst,
                                                 const float* __restrict__ xs,
                                                 const float* __restrict__ alpha,
                                                 const float* __restrict__ sbuf,
                                                 float* __restrict__ out) {
    int warp = (int)((blockIdx.x * blockDim.x + threadIdx.x) >> 5);
    int lane = threadIdx.x & 31;
    if (warp >= ETOT) return;
    int s_, d_;
    if (warp < EE) { s_ = src[warp]; d_ = dst[warp]; }
    else           { s_ = warp - EE; d_ = s_; }
    for (int seg = 0; seg < 2; ++seg) {
        int c0 = 4 * lane + 128 * seg;
        int h = c0 >> 5;
        float w = alpha[(size_t)warp * HH + h] /
                  (sbuf[(size_t)d_ * HH + h] + 1e-16f);
        float4 xv = *(const float4*)(xs + (size_t)s_ * DD + c0);
        float* op = out + (size_t)d_ * DD + c0;
        atomicAdd(op + 0, xv.x * w);
        atomicAdd(op + 1, xv.y * w);
        atomicAdd(op + 2, xv.z * w);
        atomicAdd(op + 3, xv.w * w);
    }
}

// ---------------- bias + layernorm + residual (warp per row) ----------------
__global__ __launch_bounds__(256) void k_ln_res(const float* __restrict__ T,
                                                const float* __restrict__ bo,
                                                const float* __restrict__ bvec,
                                                const float* __restrict__ lg,
                                                const float* __restrict__ lb,
                                                const float* __restrict__ resid,
                                                float* __restrict__ out) {
    int warp = (int)((blockIdx.x * blockDim.x + threadIdx.x) >> 5);
    int lane = threadIdx.x & 31;
    if (warp >= NN) return;
    const float* row = T + (size_t)warp * DD;
    float v[8];
    float s = 0.f;
    for (int j = 0; j < 8; ++j) {
        int c = lane + 32 * j;
        v[j] = row[c] + bo[c] + bvec[c];
        s += v[j];
    }
    for (int o = 16; o > 0; o >>= 1) s += __shfl_xor(s, o, 32);
    float mu = s * (1.0f / DD);
    float q = 0.f;
    for (int j = 0; j < 8; ++j) {
        float d = v[j] - mu;
        q += d * d;
    }
    for (int o = 16; o > 0; o >>= 1) q += __shfl_xor(q, o, 32);
    float rstd = rsqrtf(q * (1.0f / DD) + EPS_LN);
    for (int j = 0; j < 8; ++j) {
        int c = lane + 32 * j;
        out[(size_t)warp * DD + c] =
            (v[j] - mu) * rstd * lg[c] + lb[c] + resid[(size_t)warp * DD + c];
    }
}

// ---------------- batchnorm stats (column-wise over N) ----------------------
__global__ __launch_bounds__(256) void k_bn_stats(const float* __restrict__ X,
                                                  float* __restrict__ csum,
                                                  float* __restrict__ csumsq) {
    int col = threadIdx.x;  // 256
    float s = 0.f, q = 0.f;
    for (int r = blockIdx.x; r < NN; r += gridDim.x) {
        float v = X[(size_t)r * DD + col];
        s += v;
        q += v * v;
    }
    atomicAdd(&csum[col], s);
    atomicAdd(&csumsq[col], q);
}

// ---------------- batchnorm apply (+relu / +final residual+relu) ------------
__global__ void k_bn_apply(const float* __restrict__ X, const float* __restrict__ csum,
                           const float* __restrict__ csumsq, const float* __restrict__ g,
                           const float* __restrict__ bvec, const float* __restrict__ resid,
                           float* __restrict__ out, int do_relu, int final_relu) {
    long idx = (long)blockIdx.x * blockDim.x + threadIdx.x;
    if (idx >= (long)NN * DD) return;
    int col = (int)(idx & (DD - 1));
    float mu = csum[col] * (1.0f / NN);
    float var = csumsq[col] * (1.0f / NN) - mu * mu;
    float v = (X[idx] - mu) * rsqrtf(var + EPS_BN) * g[col] + bvec[col];
    if (do_relu) v = fmaxf(v, 0.0f);
    if (resid) v += resid[idx];
    if (final_relu) v = fmaxf(v, 0.0f);
    out[idx] = v;
}

// ---------------------------------------------------------------------------
extern "C" void kernel_launch(void* const* d_in, const int* in_sizes, int n_in,
                              void* d_out, int out_size, void* d_ws, size_t ws_size,
                              hipStream_t stream) {
    (void)in_sizes; (void)n_in; (void)out_size; (void)ws_size;

    const float* x  = (const float*)d_in[0];
    const int* eidx = (const int*)d_in[1];
    const int* src  = eidx;
    const int* dst  = eidx + EE;
    const float* ea = (const float*)d_in[2];
    const int* et   = (const int*)d_in[3];

    // per-layer parameter pointers (12 each, starting at index 4)
    const float* P[2][12];
    for (int L = 0; L < 2; ++L)
        for (int k = 0; k < 12; ++k) P[L][k] = (const float*)d_in[4 + L * 12 + k];
    // order: Ws, Wd, as, ad, We, ae, te, Wo, bo, b, lg, lb
    const float* bng1 = (const float*)d_in[28];
    const float* bnb1 = (const float*)d_in[29];
    const float* bng2 = (const float*)d_in[30];
    const float* bnb2 = (const float*)d_in[31];

    // ----- workspace carve -----
    char* w = (char*)d_ws;
    float* bufA = (float*)w; w += (size_t)NN * DD * 4;  // xs / gemm tmp
    float* bufB = (float*)w; w += (size_t)NN * DD * 4;  // accum / layer2 ln out
    float* bufC = (float*)w; w += (size_t)NN * DD * 4;  // layer1 output (h1)
    float* alphab = (float*)w; w += (size_t)ETOT * HH * 4;
    float* a_i = (float*)w; w += (size_t)NN * HH * 4;
    float* a_j = (float*)w; w += (size_t)NN * HH * 4;
    unsigned* mbuf = (unsigned*)w; w += (size_t)NN * HH * 4;
    float* sbuf = (float*)w; w += (size_t)NN * HH * 4;
    float* vWd = (float*)w; w += 2048 * 4;
    float* vWs = (float*)w; w += 2048 * 4;
    float* Me  = (float*)w; w += 128 * 4;
    float* tt  = (float*)w; w += 64 * 4;
    float* csum   = (float*)w; w += 256 * 4;
    float* csumsq = (float*)w; w += 256 * 4;

    const int ZB = 256;
    auto zero = [&](float* p, long n) {
        k_zero<<<(unsigned)((n + ZB - 1) / ZB), ZB, 0, stream>>>(p, n);
    };

    const int gemm_blocks = NN / 16;        // 1250
    const int nl_blocks   = (NN * 32 + 255) / 256;   // warp per node
    const int ea_blocks   = (ETOT + 255) / 256;
    const long ee_total   = (long)ETOT * HH;
    const int ex_blocks   = (int)((ee_total + 255) / 256);
    const int sc_blocks   = (int)(((long)ETOT * 32 + 255) / 256);
    const int ln_blocks   = (NN * 32 + 255) / 256;
    const long nd_total   = (long)NN * DD;
    const int bn_blocks   = (int)((nd_total + 255) / 256);

    auto run_layer = [&](const float* xin, const float* const* Pp,
                         const float* bng, const float* bnb,
                         float* lnout, float* bnout, const float* bn_resid,
                         int do_relu, int final_relu) {
        const float* Ws = Pp[0];  const float* Wd = Pp[1];
        const float* as_ = Pp[2]; const float* ad_ = Pp[3];
        const float* We = Pp[4];  const float* ae_ = Pp[5];
        const float* te = Pp[6];  const float* Wo = Pp[7];
        const float* bo = Pp[8];  const float* bvec = Pp[9];
        const float* lg = Pp[10]; const float* lb = Pp[11];

        // fold attention vectors into small matrices
        k_prep<<<17, 256, 0, stream>>>(Wd, as_, Ws, ad_, We, ae_, te,
                                       vWd, vWs, Me, tt);
        // xs = xin @ Ws   (WMMA)
        k_gemm256<<<gemm_blocks, 256, 0, stream>>>(xin, Ws, bufA);
        // per-node logits
        k_node_logits<<<nl_blocks, 256, 0, stream>>>(xin, vWd, vWs, a_i, a_j);
        // clear segment buffers + accumulator
        zero(bufB, nd_total);
        zero((float*)mbuf, (long)NN * HH);   // 0u == ordered(-inf)
        zero(sbuf, (long)NN * HH);
        // edge logits + segment max
        k_edge_alpha<<<ea_blocks, 256, 0, stream>>>(src, dst, ea, et, a_i, a_j,
                                                    Me, tt, alphab, mbuf);
        // exp + segment sum
        k_edge_exp<<<ex_blocks, 256, 0, stream>>>(dst, alphab, mbuf, sbuf);
        // gather + scatter-add messages
        k_scatter<<<sc_blocks, 256, 0, stream>>>(src, dst, bufA, alphab, sbuf, bufB);
        // out @ Wo   (WMMA)  -> bufA (xs dead)
        k_gemm256<<<gemm_blocks, 256, 0, stream>>>(bufB, Wo, bufA);
        // bias + LN + residual
        k_ln_res<<<ln_blocks, 256, 0, stream>>>(bufA, bo, bvec, lg, lb, xin, lnout);
        // batchnorm
        zero(csum, 512);  // csum + csumsq contiguous
        k_bn_stats<<<256, 256, 0, stream>>>(lnout, csum, csumsq);
        k_bn_apply<<<bn_blocks, 256, 0, stream>>>(lnout, csum, csumsq, bng, bnb,
                                                  bn_resid, bnout, do_relu,
                                                  final_relu);
    };

    // Layer 1: BN(relu) in place -> bufC
    run_layer(x, P[0], bng1, bnb1, /*lnout=*/bufC, /*bnout=*/bufC,
              /*bn_resid=*/nullptr, /*do_relu=*/1, /*final_relu=*/0);
    // Layer 2: BN(no relu), + original-x residual, final relu -> d_out
    run_layer(bufC, P[1], bng2, bnb2, /*lnout=*/bufB, /*bnout=*/(float*)d_out,
              /*bn_resid=*/x, /*do_relu=*/0, /*final_relu=*/1);
}